// CLAGCN_83468394431058
// MI455X (gfx1250) — compile-verified
//
#include <hip/hip_runtime.h>
#include <hip/hip_bf16.h>

typedef __attribute__((ext_vector_type(2))) float v2f;
typedef __attribute__((ext_vector_type(8))) float v8f;

#define GN 50000
#define GE 800000
#define NF 256
#define NH 128
#define NC 16

// ---------------------------------------------------------------------------
// GEMM1: C[n,128] = A[n,256] @ W[256,128], fp32 WMMA 16x16x4.
// Block = 256 threads = 8 waves; each wave: 16 rows x 128 cols (8 col tiles).
// W staged through LDS in 64-row chunks, padded stride 136 (bank-conflict-free
// across the lane 0-15 / 16-31 halves of wave32 on 64 banks).
// ---------------------------------------------------------------------------
__global__ __launch_bounds__(256) void k_gemm_nf256(
    const float* __restrict__ A, const float* __restrict__ W,
    float* __restrict__ C, int n)
{
    constexpr int WSTR = 136;                 // padded LDS row stride (floats)
    __shared__ float sW[64 * WSTR];           // 34.8 KB chunk of W
    const int lane = threadIdx.x & 31;
    const int wave = threadIdx.x >> 5;
    const int row0 = blockIdx.x * 128 + wave * 16;
    const int arow = row0 + (lane & 15);
    const int koff = (lane >> 4) * 2;         // ISA A-layout: lanes 0-15 K0,K1; 16-31 K2,K3
    const bool arow_ok = arow < n;

    v8f acc[8] = {};

    for (int kc = 0; kc < NF; kc += 64) {
        for (int i = threadIdx.x; i < 64 * NH; i += 256)
            sW[(i >> 7) * WSTR + (i & 127)] = W[(size_t)(kc + (i >> 7)) * NH + (i & 127)];
        __syncthreads();
        #pragma unroll
        for (int k0 = 0; k0 < 64; k0 += 4) {
            v2f a = {0.f, 0.f};
            if (arow_ok) {
                const float* ap = A + (size_t)arow * NF + kc + k0 + koff;
                a.x = ap[0]; a.y = ap[1];
            }
            const int krow = k0 + koff;
            #pragma unroll
            for (int t = 0; t < 8; ++t) {
                const int col = t * 16 + (lane & 15);
                v2f b;
                b.x = sW[krow * WSTR + col];
                b.y = sW[(krow + 1) * WSTR + col];
                acc[t] = __builtin_amdgcn_wmma_f32_16x16x4_f32(
                    false, a, false, b, (short)0, acc[t], false, false);
            }
        }
        __syncthreads();
    }
    #pragma unroll
    for (int t = 0; t < 8; ++t) {
        const int col = t * 16 + (lane & 15);
        #pragma unroll
        for (int v = 0; v < 8; ++v) {
            const int r = row0 + v + ((lane >> 4) << 3);  // C layout: lanes>=16 -> M=v+8
            if (r < n) C[(size_t)r * NH + col] = acc[t][v];
        }
    }
}

// ---------------------------------------------------------------------------
// GEMM2: P[n,16] = (w1*H1 + w2*H2)[n,256] @ W2[256,16]. Gate fused into A load.
// ---------------------------------------------------------------------------
__global__ __launch_bounds__(256) void k_gemm2(
    const float* __restrict__ H1, const float* __restrict__ H2,
    const float* __restrict__ gate, const float* __restrict__ W2,
    float* __restrict__ P, int n)
{
    __shared__ float sW[NF * NC];             // 16 KB, whole W2
    const float w1 = gate[0], w2 = gate[1];
    for (int i = threadIdx.x; i < NF * NC; i += 256) sW[i] = W2[i];
    __syncthreads();

    const int lane = threadIdx.x & 31;
    const int wave = threadIdx.x >> 5;
    const int row0 = blockIdx.x * 128 + wave * 16;
    const int arow = row0 + (lane & 15);
    const int koff = (lane >> 4) * 2;
    const bool arow_ok = arow < n;
    const int col = lane & 15;

    v8f acc = {};
    #pragma unroll 4
    for (int k0 = 0; k0 < NF; k0 += 4) {
        v2f a = {0.f, 0.f};
        if (arow_ok) {
            const size_t idx = (size_t)arow * NF + k0 + koff;
            a.x = w1 * H1[idx]     + w2 * H2[idx];
            a.y = w1 * H1[idx + 1] + w2 * H2[idx + 1];
        }
        const int krow = k0 + koff;
        v2f b;
        b.x = sW[krow * NC + col];
        b.y = sW[(krow + 1) * NC + col];
        acc = __builtin_amdgcn_wmma_f32_16x16x4_f32(
            false, a, false, b, (short)0, acc, false, false);
    }
    #pragma unroll
    for (int v = 0; v < 8; ++v) {
        const int r = row0 + v + ((lane >> 4) << 3);
        if (r < n) P[(size_t)r * NC + col] = acc[v];
    }
}

// ---------------------------------------------------------------------------
// SpMM (128 feats): one wave per edge; lane handles 4 consecutive feats.
// Accumulates via f32 L2 atomics into H (row stride hstride, col offset in base).
// ---------------------------------------------------------------------------
__global__ __launch_bounds__(256) void k_spmm128(
    const int* __restrict__ row, const int* __restrict__ col,
    const float* __restrict__ val, const float* __restrict__ S,
    float* __restrict__ H, int hstride, int e)
{
    const int eidx = (int)((blockIdx.x * 256u + threadIdx.x) >> 5);
    const int lane = threadIdx.x & 31;
    if (eidx >= e) return;
    const int r = row[eidx], c = col[eidx];
    const float w = val[eidx];
    const float4 v = ((const float4*)(S + (size_t)c * NH))[lane];
    float* dst = H + (size_t)r * hstride + lane * 4;
    atomicAdd(dst + 0, w * v.x);
    atomicAdd(dst + 1, w * v.y);
    atomicAdd(dst + 2, w * v.z);
    atomicAdd(dst + 3, w * v.w);
}

// SpMM (16 feats): 16 threads per edge.
__global__ __launch_bounds__(256) void k_spmm16(
    const int* __restrict__ row, const int* __restrict__ col,
    const float* __restrict__ val, const float* __restrict__ P,
    float* __restrict__ OUT, int e)
{
    const long long t = (long long)blockIdx.x * 256 + threadIdx.x;
    const int eidx = (int)(t >> 4);
    const int f = (int)(t & 15);
    if (eidx >= e) return;
    const int r = row[eidx], c = col[eidx];
    atomicAdd(&OUT[(size_t)r * NC + f], val[eidx] * P[(size_t)c * NC + f]);
}

// In-place bias + ReLU over a [n,128] column block of a stride-256 matrix.
__global__ __launch_bounds__(256) void k_biasrelu(
    float* __restrict__ H, const float* __restrict__ b, int n)
{
    const int idx = blockIdx.x * 256 + threadIdx.x;
    if (idx >= n * NH) return;
    const int r = idx >> 7, c = idx & 127;
    float* p = H + (size_t)r * NF + c;
    const float v = *p + b[c];
    *p = v > 0.f ? v : 0.f;
}

// Bias add over the two contiguous gnn_prop regions (2*n*16 elems, col = i&15).
__global__ __launch_bounds__(256) void k_bias16(
    float* __restrict__ Pq, const float* __restrict__ b, int total)
{
    const int i = blockIdx.x * 256 + threadIdx.x;
    if (i < total) Pq[i] += b[i & 15];
}

__device__ __forceinline__ float sigmoidf_(float x) {
    return 1.f / (1.f + __expf(-x));
}

// Gate on last row of h1/h2 (256-dim dot + sigmoid + L1 normalize).
__global__ __launch_bounds__(256) void k_gate1(
    const float* __restrict__ h1last, const float* __restrict__ h2last,
    const float* __restrict__ wa, const float* __restrict__ ba,
    const float* __restrict__ wb, const float* __restrict__ bb,
    float* __restrict__ gate)
{
    __shared__ float s1[256], s2[256];
    const int t = threadIdx.x;
    s1[t] = h1last[t] * wa[t];
    s2[t] = h2last[t] * wb[t];
    __syncthreads();
    for (int off = 128; off; off >>= 1) {
        if (t < off) { s1[t] += s1[t + off]; s2[t] += s2[t + off]; }
        __syncthreads();
    }
    if (t == 0) {
        const float la = sigmoidf_(s1[0] + ba[0]);
        const float lb = sigmoidf_(s2[0] + bb[0]);
        const float s = fmaxf(la + lb, 1e-12f);
        gate[0] = la / s;
        gate[1] = lb / s;
    }
}

// Gate on last row of gnn_prop1/2 (16-dim dot).
__global__ __launch_bounds__(32) void k_gate2(
    const float* __restrict__ p1last, const float* __restrict__ p2last,
    const float* __restrict__ wa, const float* __restrict__ ba,
    const float* __restrict__ wb, const float* __restrict__ bb,
    float* __restrict__ gate)
{
    if (threadIdx.x == 0) {
        float d1 = 0.f, d2 = 0.f;
        #pragma unroll
        for (int i = 0; i < NC; ++i) {
            d1 += p1last[i] * wa[i];
            d2 += p2last[i] * wb[i];
        }
        const float la = sigmoidf_(d1 + ba[0]);
        const float lb = sigmoidf_(d2 + bb[0]);
        const float s = fmaxf(la + lb, 1e-12f);
        gate[2] = la / s;
        gate[3] = lb / s;
    }
}

__global__ __launch_bounds__(256) void k_combine(
    const float* __restrict__ p1, const float* __restrict__ p2,
    const float* __restrict__ gate, float* __restrict__ out, int total)
{
    const int i = blockIdx.x * 256 + threadIdx.x;
    if (i < total) out[i] = gate[2] * p1[i] + gate[3] * p2[i];
}

extern "C" void kernel_launch(void* const* d_in, const int* in_sizes, int n_in,
                              void* d_out, int out_size, void* d_ws, size_t ws_size,
                              hipStream_t stream)
{
    const float* x1a = (const float*)d_in[0];
    const float* x1b = (const float*)d_in[1];
    const float* x2a = (const float*)d_in[2];
    const float* x2b = (const float*)d_in[3];
    const int*   row1 = (const int*)d_in[4];
    const int*   col1 = (const int*)d_in[5];
    const float* ew1 = (const float*)d_in[6];
    const int*   row2 = (const int*)d_in[7];
    const int*   col2 = (const int*)d_in[8];
    const float* ew2 = (const float*)d_in[9];
    const float* W1a = (const float*)d_in[10];
    const float* b1a = (const float*)d_in[11];
    const float* W1b = (const float*)d_in[12];
    const float* b1b = (const float*)d_in[13];
    const float* W2  = (const float*)d_in[14];
    const float* b2  = (const float*)d_in[15];
    const float* fw1_w = (const float*)d_in[16];
    const float* fw1_b = (const float*)d_in[17];
    const float* fw2_w = (const float*)d_in[18];
    const float* fw2_b = (const float*)d_in[19];
    const float* g1_w = (const float*)d_in[20];
    const float* g1_b = (const float*)d_in[21];
    const float* g2_w = (const float*)d_in[22];
    const float* g2_b = (const float*)d_in[23];

    const int n = GN, e = GE;

    // Workspace layout (floats)
    float* S    = (float*)d_ws;                    // [n,128] support scratch (reused 4x)
    float* h1   = S  + (size_t)n * NH;             // [n,256]
    float* h2   = h1 + (size_t)n * 2 * NH;         // [n,256]
    float* P    = h2 + (size_t)n * 2 * NH;         // [n,16]
    float* gate = P  + (size_t)n * NC;             // [4]: w1,w2,a1,a2

    float* out = (float*)d_out;                    // [n,16]
    float* p1  = out + (size_t)n * NC;             // gnn_prop1 [n,16]
    float* p2  = out + 2 * (size_t)n * NC;         // gnn_prop2 [n,16]

    // Zero accumulation targets (h1,h2 contiguous; p1,p2 contiguous).
    hipMemsetAsync(h1, 0, (size_t)n * 2 * NH * sizeof(float) * 2, stream);
    hipMemsetAsync(p1, 0, (size_t)n * NC * sizeof(float) * 2, stream);

    const dim3 blk(256);
    const int gemmB  = (n + 127) / 128;
    const int spmmB  = (e + 7) / 8;                    // 8 waves/block, 1 edge/wave
    const int brB    = (n * NH + 255) / 256;
    const int spm16B = (int)(((long long)e * NC + 255) / 256);

    // --- Layer 1: four GCN branches, S buffer reused sequentially ---
    k_gemm_nf256<<<gemmB, blk, 0, stream>>>(x1a, W1a, S, n);
    k_spmm128<<<spmmB, blk, 0, stream>>>(row1, col1, ew1, S, h1, NF, e);
    k_biasrelu<<<brB, blk, 0, stream>>>(h1, b1a, n);

    k_gemm_nf256<<<gemmB, blk, 0, stream>>>(x1b, W1b, S, n);
    k_spmm128<<<spmmB, blk, 0, stream>>>(row1, col1, ew1, S, h1 + NH, NF, e);
    k_biasrelu<<<brB, blk, 0, stream>>>(h1 + NH, b1b, n);

    k_gemm_nf256<<<gemmB, blk, 0, stream>>>(x2a, W1a, S, n);
    k_spmm128<<<spmmB, blk, 0, stream>>>(row2, col2, ew2, S, h2, NF, e);
    k_biasrelu<<<brB, blk, 0, stream>>>(h2, b1a, n);

    k_gemm_nf256<<<gemmB, blk, 0, stream>>>(x2b, W1b, S, n);
    k_spmm128<<<spmmB, blk, 0, stream>>>(row2, col2, ew2, S, h2 + NH, NF, e);
    k_biasrelu<<<brB, blk, 0, stream>>>(h2 + NH, b1b, n);

    // --- Gate 1 (last node only) + fused combine inside GEMM2 ---
    k_gate1<<<1, 256, 0, stream>>>(h1 + (size_t)(n - 1) * NF, h2 + (size_t)(n - 1) * NF,
                                   fw1_w, fw1_b, fw2_w, fw2_b, gate);
    k_gemm2<<<gemmB, blk, 0, stream>>>(h1, h2, gate, W2, P, n);

    // --- Layer 2 SpMMs into output regions ---
    k_spmm16<<<spm16B, blk, 0, stream>>>(row1, col1, ew1, P, p1, e);
    k_spmm16<<<spm16B, blk, 0, stream>>>(row2, col2, ew2, P, p2, e);
    k_bias16<<<(2 * n * NC + 255) / 256, blk, 0, stream>>>(p1, b2, 2 * n * NC);

    // --- Gate 2 + final combine ---
    k_gate2<<<1, 32, 0, stream>>>(p1 + (size_t)(n - 1) * NC, p2 + (size_t)(n - 1) * NC,
                                  g1_w, g1_b, g2_w, g2_b, gate);
    k_combine<<<(n * NC + 255) / 256, blk, 0, stream>>>(p1, p2, gate, out, n * NC);
}